// RotationGate_36000415875235
// MI455X (gfx1250) — compile-verified
//
#include <hip/hip_runtime.h>
#include <math.h>

// ---------------------------------------------------------------------------
// RX(theta) on qubit 0 of a 22-qubit state, batch=16, f32.
// Memory-bound streaming kernel (AI ~0.25 FLOP/B; ~1 GB traffic -> ~44us at
// 23.3 TB/s). CDNA5 path exercised: per-lane async global->LDS b128 copies
// (ASYNCcnt) with double buffering, ds_load_b128 consumption, and
// non-temporal b128 stores (1 GB stream >> 192 MB L2).
// ---------------------------------------------------------------------------

typedef float v4f __attribute__((ext_vector_type(4)));
typedef int   v4i_t __attribute__((vector_size(16)));  // matches builtin proto

constexpr int    BATCH = 16;
constexpr int    TPB   = 256;
constexpr int    ITERS = 4;
constexpr size_t DIMQ  = (size_t)1 << 22;      // 4194304 amplitudes
constexpr size_t HALFQ = DIMQ >> 1;            // rows per half (j=0 / j=1)
constexpr size_t WORK  = HALFQ * (BATCH / 4);  // 8,388,608 float4 work items
constexpr size_t HALF4 = HALFQ * 4;            // v4f offset: j=0 row -> j=1 row
constexpr size_t DIM4  = DIMQ * 4;             // v4f offset: real plane -> imag plane

static_assert(WORK % (TPB * ITERS) == 0, "grid must divide evenly");
constexpr unsigned NBLOCKS = (unsigned)(WORK / (TPB * ITERS));  // 8192

#if __has_builtin(__builtin_amdgcn_global_load_async_to_lds_b128)
#define USE_ASYNC_LDS 1
#else
#define USE_ASYNC_LDS 0
#endif

#if USE_ASYNC_LDS
__device__ __forceinline__ void gl_async_b128(const v4f* g, v4f* l) {
  // global_load_async_to_lds_b128: per-lane 16B global->LDS, tracked by ASYNCcnt
  __builtin_amdgcn_global_load_async_to_lds_b128(
      (__attribute__((address_space(1))) v4i_t*)g,
      (__attribute__((address_space(3))) v4i_t*)l,
      /*offset=*/0, /*cpol=*/0);
}

template <int N>
__device__ __forceinline__ void wait_async() {
#if __has_builtin(__builtin_amdgcn_s_wait_asynccnt)
  __builtin_amdgcn_s_wait_asynccnt(N);
#else
  asm volatile("s_wait_asynccnt %0" ::"n"(N) : "memory");
#endif
}
#endif  // USE_ASYNC_LDS

__global__ __launch_bounds__(TPB) void rotx22_kernel(
    const float* __restrict__ theta, const float* __restrict__ sre,
    const float* __restrict__ sim, float* __restrict__ out) {
  const int tid = threadIdx.x;

  // Per-block gate coefficients: c = cos(theta/2), s = sin(theta/2).
  __shared__ __align__(16) float csh[BATCH];
  __shared__ __align__(16) float ssh[BATCH];
  if (tid < BATCH) {
    const float t = 0.5f * theta[tid];
    csh[tid] = cosf(t);
    ssh[tid] = sinf(t);
  }
  __syncthreads();

  const int b4 = tid & 3;  // which group of 4 batch columns this thread owns
  const v4f c = *reinterpret_cast<const v4f*>(csh + 4 * b4);
  const v4f s = *reinterpret_cast<const v4f*>(ssh + 4 * b4);

  const v4f* re = reinterpret_cast<const v4f*>(sre);
  const v4f* im = reinterpret_cast<const v4f*>(sim);
  v4f* o = reinterpret_cast<v4f*>(out);

  // Work item w: float offset of re0 stream is exactly 4*w (fully contiguous).
  const size_t base = (size_t)blockIdx.x * (TPB * ITERS) + (size_t)tid;

#if USE_ASYNC_LDS
  __shared__ v4f tile[2][4][TPB];  // [buf][{re0,im0,re1,im1}][lane] = 32 KB

  auto issue = [&](size_t w, int buf) {
    gl_async_b128(re + w,         &tile[buf][0][tid]);
    gl_async_b128(im + w,         &tile[buf][1][tid]);
    gl_async_b128(re + w + HALF4, &tile[buf][2][tid]);
    gl_async_b128(im + w + HALF4, &tile[buf][3][tid]);
  };

  issue(base, 0);  // prologue: prefetch tile 0
#pragma unroll
  for (int i = 0; i < ITERS; ++i) {
    const size_t w = base + (size_t)i * TPB;
    const int buf = i & 1;
    if (i + 1 < ITERS) {
      issue(w + TPB, buf ^ 1);  // prefetch next tile into other buffer
      wait_async<4>();          // tile i's 4 copies have landed in LDS
    } else {
      wait_async<0>();
    }
    const v4f re0 = tile[buf][0][tid];
    const v4f im0 = tile[buf][1][tid];
    const v4f re1 = tile[buf][2][tid];
    const v4f im1 = tile[buf][3][tid];

    const v4f o0r = c * re0 + s * im1;
    const v4f o0i = c * im0 - s * re1;
    const v4f o1r = c * re1 + s * im0;
    const v4f o1i = c * im1 - s * re0;

    __builtin_nontemporal_store(o0r, o + w);
    __builtin_nontemporal_store(o1r, o + w + HALF4);
    __builtin_nontemporal_store(o0i, o + w + DIM4);
    __builtin_nontemporal_store(o1i, o + w + DIM4 + HALF4);
  }
#else
  // Fallback: direct non-temporal b128 streaming (no async builtins found).
#pragma unroll
  for (int i = 0; i < ITERS; ++i) {
    const size_t w = base + (size_t)i * TPB;
    const v4f re0 = __builtin_nontemporal_load(re + w);
    const v4f im0 = __builtin_nontemporal_load(im + w);
    const v4f re1 = __builtin_nontemporal_load(re + w + HALF4);
    const v4f im1 = __builtin_nontemporal_load(im + w + HALF4);

    const v4f o0r = c * re0 + s * im1;
    const v4f o0i = c * im0 - s * re1;
    const v4f o1r = c * re1 + s * im0;
    const v4f o1i = c * im1 - s * re0;

    __builtin_nontemporal_store(o0r, o + w);
    __builtin_nontemporal_store(o1r, o + w + HALF4);
    __builtin_nontemporal_store(o0i, o + w + DIM4);
    __builtin_nontemporal_store(o1i, o + w + DIM4 + HALF4);
  }
#endif
}

extern "C" void kernel_launch(void* const* d_in, const int* in_sizes, int n_in,
                              void* d_out, int out_size, void* d_ws,
                              size_t ws_size, hipStream_t stream) {
  (void)in_sizes; (void)n_in; (void)out_size; (void)d_ws; (void)ws_size;
  const float* theta = (const float*)d_in[0];   // (16,)
  const float* sre   = (const float*)d_in[1];   // (DIM, 16)
  const float* sim   = (const float*)d_in[2];   // (DIM, 16)
  float* out = (float*)d_out;                   // (2, DIM, 16)
  rotx22_kernel<<<dim3(NBLOCKS), dim3(TPB), 0, stream>>>(theta, sre, sim, out);
}